// GCN_NC_43542378447162
// MI455X (gfx1250) — compile-verified
//
#include <hip/hip_runtime.h>
#include <cmath>

typedef __attribute__((ext_vector_type(2))) float v2f;
typedef __attribute__((ext_vector_type(8))) float v8f;

#define NFEAT 128   // input & hidden feature dim (reference constant)
#define NCLS  40    // output classes (reference constant)
#define LD2   48    // padded leading dim for layer-2 buffers (3 x 16 tiles)
#define KDIM  128   // contraction dim for both layers
#define WPB   8     // waves per GEMM block (256 threads)

// ---------------------------------------------------------------- utilities

__global__ void fill_f32(float* __restrict__ p, long long n, float v) {
  long long i = blockIdx.x * (long long)blockDim.x + threadIdx.x;
  long long stride = (long long)gridDim.x * blockDim.x;
  for (; i < n; i += stride) p[i] = v;
}

__global__ void degree_kernel(const int* __restrict__ col, float* __restrict__ deg, int E) {
  int e = blockIdx.x * blockDim.x + threadIdx.x;
  if (e < E) atomicAdd(&deg[col[e]], 1.0f);
}

__global__ void rsqrt_kernel(float* __restrict__ deg, int N) {
  int i = blockIdx.x * blockDim.x + threadIdx.x;
  if (i < N) {
    float d = deg[i];
    deg[i] = (d > 0.0f) ? rsqrtf(d) : 0.0f;
  }
}

// ------------------------------------------------- fp32 WMMA tiled GEMM
// C[M x ldc] = A[M x KDIM] * B[KDIM x Nvalid], row-major.
// One block = 8 waves sharing one 16-col B tile staged in LDS (padded with
// zeros past Nvalid at fill time, so the hot loop has no divergent guards).
// Each wave owns one 16x16 C tile; K-loop steps by 4 with
// V_WMMA_F32_16X16X4_F32.  EXEC is all-ones for every live wave at the WMMA.
__global__ __launch_bounds__(256) void gemm_wmma_f32(
    const float* __restrict__ A, const float* __restrict__ B, float* __restrict__ C,
    int M, int Nvalid, int Ntiles, int ldc) {
  __shared__ float Bs[KDIM * 16];  // 8 KB

  int nt = blockIdx.x % Ntiles;
  int mtBase = (blockIdx.x / Ntiles) * WPB;
  int n0 = nt << 4;
  int t = threadIdx.x;

  // Cooperative B-tile fill: 2048 elements / 256 threads = 8 each, coalesced.
#pragma unroll
  for (int j = 0; j < 8; ++j) {
    int idx = t + j * 256;
    int kb = idx >> 4;
    int n = n0 + (idx & 15);
    Bs[idx] = (n < Nvalid) ? B[(long long)kb * Nvalid + n] : 0.0f;
  }
  __syncthreads();

  int w = t >> 5;
  int mt = mtBase + w;
  int Mtiles = (M + 15) >> 4;
  if (mt >= Mtiles) return;   // wave-uniform exit (after the barrier)

  int lane = t & 31;
  int l16 = lane & 15;
  int hi  = lane >> 4;        // 0: lanes 0-15, 1: lanes 16-31
  int kh  = hi << 1;          // K sub-offset 0 or 2 (A & B WMMA layouts)

  int m0 = mt << 4;
  int m = m0 + l16;           // A row this lane feeds (A is 16x4: lane->M)
  if (m >= M) m = M - 1;      // clamp loads; stores guarded below
  const float* Arow = A + (long long)m * KDIM;

  v8f acc = {};
#pragma unroll 8
  for (int k = 0; k < KDIM; k += 4) {
    v2f a = *(const v2f*)(Arow + k + kh);          // global_load_b64
    v2f b;
    b.x = Bs[(k + kh) * 16 + l16];                 // ds_load_2addr, no conflicts
    b.y = Bs[(k + kh + 1) * 16 + l16];
    acc = __builtin_amdgcn_wmma_f32_16x16x4_f32(
        /*neg_a=*/false, a, /*neg_b=*/false, b,
        /*c_mod=*/(short)0, acc, /*reuse_a=*/false, /*reuse_b=*/false);
  }

  // D 16x16 layout: lane -> col N=l16; VGPR r -> row M = r + hi*8
  int mbase = m0 + hi * 8;
  float* Cout = C + (long long)mbase * ldc + n0 + l16;
  if (m0 + 16 <= M) {
    // Full tile (always taken for M=50000): unguarded stores, no EXEC churn.
#pragma unroll
    for (int r = 0; r < 8; ++r) Cout[(long long)r * ldc] = acc[r];
  } else {
    // Generic tail tile.
#pragma unroll
    for (int r = 0; r < 8; ++r)
      if (mbase + r < M) Cout[(long long)r * ldc] = acc[r];
  }
}

// ------------------------------------------- edge gather + scatter-add
// One thread handles 4 consecutive features of one edge: b128 gather +
// 4x global_atomic_add_f32.  F4 = F/4.
__global__ void scatter_add4(const int* __restrict__ row, const int* __restrict__ col,
                             const float* __restrict__ dinv, const float* __restrict__ h,
                             float* __restrict__ agg, int E, int F4, int ld) {
  long long total = (long long)E * F4;
  long long i = blockIdx.x * (long long)blockDim.x + threadIdx.x;
  long long stride = (long long)gridDim.x * blockDim.x;
  for (; i < total; i += stride) {
    int e = (int)(i / F4);
    int f = (int)(i - (long long)e * F4) << 2;
    int r = row[e];
    int c = col[e];
    float nrm = dinv[r] * dinv[c];
    const float4 hv = *(const float4*)(h + (long long)r * ld + f);
    float* a = agg + (long long)c * ld + f;
    atomicAdd(a + 0, hv.x * nrm);
    atomicAdd(a + 1, hv.y * nrm);
    atomicAdd(a + 2, hv.z * nrm);
    atomicAdd(a + 3, hv.w * nrm);
  }
}

// ----------------------- layer-1 epilogue: self-loop + bias + ReLU (in place)
__global__ void bias_relu_self(const float* __restrict__ h, const float* __restrict__ dinv,
                               const float* __restrict__ bias, float* __restrict__ agg,
                               int Nnodes) {
  long long total = (long long)Nnodes * NFEAT;
  long long i = blockIdx.x * (long long)blockDim.x + threadIdx.x;
  long long stride = (long long)gridDim.x * blockDim.x;
  for (; i < total; i += stride) {
    int nidx = (int)(i >> 7);          // NFEAT == 128
    int f = (int)(i & (NFEAT - 1));
    float d = dinv[nidx];
    float v = agg[i] + h[i] * d * d + bias[f];
    agg[i] = v > 0.0f ? v : 0.0f;
  }
}

// ----------------------- layer-2 epilogue: self-loop + bias + log_softmax
__global__ void out_logsoftmax(const float* __restrict__ h2, const float* __restrict__ agg2,
                               const float* __restrict__ dinv, const float* __restrict__ b2,
                               float* __restrict__ out, int Nnodes) {
  int i = blockIdx.x * blockDim.x + threadIdx.x;
  if (i >= Nnodes) return;
  float d = dinv[i];
  float d2 = d * d;
  float logit[NCLS];
  float mx = -3.4e38f;
#pragma unroll
  for (int f = 0; f < NCLS; ++f) {
    float v = agg2[(long long)i * LD2 + f] + h2[(long long)i * LD2 + f] * d2 + b2[f];
    logit[f] = v;
    mx = fmaxf(mx, v);
  }
  float s = 0.0f;
#pragma unroll
  for (int f = 0; f < NCLS; ++f) s += __expf(logit[f] - mx);
  float lse = mx + __logf(s);
#pragma unroll
  for (int f = 0; f < NCLS; ++f) out[(long long)i * NCLS + f] = logit[f] - lse;
}

// ---------------------------------------------------------------- launcher

extern "C" void kernel_launch(void* const* d_in, const int* in_sizes, int n_in,
                              void* d_out, int out_size, void* d_ws, size_t ws_size,
                              hipStream_t stream) {
  const float* x  = (const float*)d_in[0];
  const int*   ei = (const int*)d_in[1];   // [2, E] flattened: row then col
  const float* W1 = (const float*)d_in[2];
  const float* b1 = (const float*)d_in[3];
  const float* W2 = (const float*)d_in[4];
  const float* b2 = (const float*)d_in[5];
  float* out = (float*)d_out;

  const int N = in_sizes[0] / NFEAT;   // 50000
  const int E = in_sizes[1] / 2;       // 800000
  const int* erow = ei;
  const int* ecol = ei + E;

  // workspace partition (floats)
  float* dinv = (float*)d_ws;                       // N       (deg -> dinv)
  float* h1   = dinv + N;                           // N * 128
  float* agg1 = h1 + (long long)N * NFEAT;          // N * 128 (-> activations)
  float* h2   = agg1 + (long long)N * NFEAT;        // N * 48
  float* agg2 = h2 + (long long)N * LD2;            // N * 48

  const int TPB = 256;
  auto blocks = [](long long n, int tpb) {
    long long b = (n + tpb - 1) / tpb;
    return (unsigned)(b > 1048576 ? 1048576 : b);
  };

  // 1) init: deg = 1.0 (self loop), zero accumulators (every call: deterministic)
  fill_f32<<<blocks(N, TPB), TPB, 0, stream>>>(dinv, N, 1.0f);
  fill_f32<<<blocks((long long)N * NFEAT, TPB), TPB, 0, stream>>>(agg1, (long long)N * NFEAT, 0.0f);
  fill_f32<<<blocks((long long)N * LD2, TPB), TPB, 0, stream>>>(agg2, (long long)N * LD2, 0.0f);

  // 2) degree + D^{-1/2}
  degree_kernel<<<(E + TPB - 1) / TPB, TPB, 0, stream>>>(ecol, dinv, E);
  rsqrt_kernel<<<(N + TPB - 1) / TPB, TPB, 0, stream>>>(dinv, N);

  const int Mtiles = (N + 15) / 16;
  const int mGroups = (Mtiles + WPB - 1) / WPB;

  // 3) h1 = x @ W1  (fp32 WMMA, B tile in LDS)
  {
    int Ntiles = NFEAT / 16;  // 8
    gemm_wmma_f32<<<mGroups * Ntiles, 256, 0, stream>>>(x, W1, h1, N, NFEAT, Ntiles, NFEAT);
  }

  // 4) edge scatter, layer 1 (float4 gather + 4 atomics)
  scatter_add4<<<blocks((long long)E * (NFEAT / 4), TPB), TPB, 0, stream>>>(
      erow, ecol, dinv, h1, agg1, E, NFEAT / 4, NFEAT);

  // 5) self-loop + bias + ReLU (agg1 becomes activations)
  bias_relu_self<<<blocks((long long)N * NFEAT, TPB), TPB, 0, stream>>>(
      h1, dinv, b1, agg1, N);

  // 6) h2 = act @ W2  (fp32 WMMA, N=40 padded to 3x16 tiles, ld=48)
  {
    int Ntiles = LD2 / 16;  // 3
    gemm_wmma_f32<<<mGroups * Ntiles, 256, 0, stream>>>(agg1, W2, h2, N, NCLS, Ntiles, LD2);
  }

  // 7) edge scatter, layer 2
  scatter_add4<<<blocks((long long)E * (NCLS / 4), TPB), TPB, 0, stream>>>(
      erow, ecol, dinv, h2, agg2, E, NCLS / 4, LD2);

  // 8) self-loop + bias + log_softmax -> output
  out_logsoftmax<<<(N + TPB - 1) / TPB, TPB, 0, stream>>>(
      h2, agg2, dinv, b2, out, N);
}